// LReLuTorch_81793357185895
// MI455X (gfx1250) — compile-verified
//
#include <hip/hip_runtime.h>

// B=8, IN=256, OUT=128, C=64, NHWC fp32.
// out = Down2H(Down2W( Down2H(Down2W( lrelu( Up2H(Up2W(x)) ) )) )) + bias
//
// K1: fused up2x + lrelu + down2x (256 -> 256), pure VALU, register strips.
// K2: 256 -> 128 separable downsample as banded GEMMs on the WMMA f32 engine.
//     Input tile is DMA'd into LDS with TENSOR_LOAD_TO_LDS (TDM, ISA ch.8),
//     W-pass result staged in LDS; 288 KB LDS total (CDNA5: 320 KB/WGP).

#define BATCH 8
#define INS 256
#define UPS 512
#define OUTS 128
#define CH 64

typedef __attribute__((ext_vector_type(2))) float v2f;
typedef __attribute__((ext_vector_type(8))) float v8f;
typedef __attribute__((ext_vector_type(4))) unsigned int u32x4;
typedef __attribute__((ext_vector_type(8))) int i32x8;
typedef __attribute__((ext_vector_type(4))) int i32x4;

__device__ __forceinline__ int iclamp(int v, int lo, int hi) {
  return v < lo ? lo : (v > hi ? hi : v);
}
__device__ __forceinline__ float lrelu(float v) { return v >= 0.0f ? v : 0.01f * v; }

// Antialiased bilinear 2x-downsample weight (JAX semantics):
// output o taps input i = 2o-1+t, t=0..3, base [1,3,3,1]/8, edges renormalized.
__device__ __forceinline__ float wdown(int o, int i, int nout, int nin) {
  int t = i - (2 * o - 1);
  float w = (t == 0 || t == 3) ? 0.125f : 0.375f;
  bool ok = (t >= 0) & (t <= 3) & (i >= 0) & (i < nin);
  w = ok ? w : 0.0f;
  if (o == 0 || o == nout - 1) w *= (8.0f / 7.0f);
  return w;
}

// ---------------------------------------------------------------------------
// K1: inter = Down2( lrelu( Up2(x) ) )  (256x256 -> 256x256), unchanged.
// ---------------------------------------------------------------------------
__global__ __launch_bounds__(256) void k1_up_lrelu_down(const float* __restrict__ x,
                                                        float* __restrict__ inter) {
  int n = blockIdx.x * 256 + threadIdx.x;
  int c = n & (CH - 1);
  int j = (n >> 6) & (INS - 1);
  int is = (n >> 14) & 63;
  int b = n >> 20;
  int i0 = is * 4;

  const float* xb = x + (size_t)b * INS * INS * CH;

  float X[6][3];
#pragma unroll
  for (int r = 0; r < 6; ++r) {
    int gr = iclamp(i0 - 1 + r, 0, INS - 1);
#pragma unroll
    for (int cc = 0; cc < 3; ++cc) {
      int gc = iclamp(j - 1 + cc, 0, INS - 1);
      X[r][cc] = xb[((size_t)gr * INS + gc) * CH + c];
    }
  }

  float V[10][3];
#pragma unroll
  for (int pl = 0; pl < 10; ++pl) {
    float w0 = (pl & 1) ? 0.25f : 0.75f;
    int l0 = pl >> 1;
#pragma unroll
    for (int cc = 0; cc < 3; ++cc)
      V[pl][cc] = w0 * X[l0][cc] + (1.0f - w0) * X[l0 + 1][cc];
  }

  float U[10][4];
#pragma unroll
  for (int pl = 0; pl < 10; ++pl) {
#pragma unroll
    for (int ql = 0; ql < 4; ++ql) {
      float w0 = (ql & 1) ? 0.25f : 0.75f;
      int l0 = ql >> 1;
      U[pl][ql] = lrelu(w0 * V[pl][l0] + (1.0f - w0) * V[pl][l0 + 1]);
    }
  }

  float hwd[4];
#pragma unroll
  for (int s = 0; s < 4; ++s) hwd[s] = wdown(j, 2 * j - 1 + s, INS, UPS);

  float* ob = inter + (size_t)b * INS * INS * CH;
#pragma unroll
  for (int il = 0; il < 4; ++il) {
    int i = i0 + il;
    float vw[4];
#pragma unroll
    for (int t = 0; t < 4; ++t) vw[t] = wdown(i, 2 * i - 1 + t, INS, UPS);
    float acc = 0.0f;
#pragma unroll
    for (int s = 0; s < 4; ++s) {
      float col = vw[0] * U[2 * il + 0][s] + vw[1] * U[2 * il + 1][s] +
                  vw[2] * U[2 * il + 2][s] + vw[3] * U[2 * il + 3][s];
      acc += hwd[s] * col;
    }
    ob[((size_t)i * INS + j) * CH + c] = acc;
  }
}

// ---------------------------------------------------------------------------
// K2: out = Down2H(Down2W(inter)) + bias via V_WMMA_F32_16X16X4_F32.
// One block = (batch, 16x16 output tile, 32-channel half), 8 waves.
//
// TDM loads I window [hi_start..+35][wi_start..+35][cb*32..+31] into LDS with
// padded 40-float lines (pad_interval=4 -> every 32 DWORDs, pad_amount=7 ->
// +8 DWORDs) so the two half-wave K-groups hit disjoint LDS bank sets.
// Window origin is clamped in-bounds; taps outside the window carry A==0.
// ---------------------------------------------------------------------------
#define IWIN 36    // hi/wi window rows/cols
#define ILINE 40   // floats per (hi,wi) line in LDS (32 data + 8 pad)
#define TWSTR 34   // T wo stride (floats): 8*34 % 64 == 16 -> disjoint banks
#define THSTR (16 * TWSTR + 16)  // 560: 2*560 % 64 == 32 -> disjoint banks
#define ITILE_FLOATS (IWIN * IWIN * ILINE)        // 51,840
#define LDS_FLOATS (ITILE_FLOATS + IWIN * THSTR)  // 72,000 -> 288,000 B

__global__ __launch_bounds__(256) void k2_down2_wmma(const float* __restrict__ inter,
                                                     const float* __restrict__ bias,
                                                     float* __restrict__ out) {
  extern __shared__ float lds_f[];
  float* itile = lds_f;                 // [IWIN hi][IWIN wi][ILINE]
  float* tbuf = lds_f + ITILE_FLOATS;   // [IWIN hi_l][16 wo * TWSTR + pad]

  int blk = blockIdx.x;
  int woT = blk & 7;
  int hoT = (blk >> 3) & 7;
  int cb = (blk >> 6) & 1;  // 32-channel half
  int b = blk >> 7;
  int wo0 = woT * 16;
  int ho0 = hoT * 16;

  int lane = threadIdx.x & 31;
  int wave = threadIdx.x >> 5;
  int ln = lane & 15;   // A: M row; B/D: N column
  int grp = lane >> 4;  // K pair {0,1} vs {2,3}

  // clamp window origin fully in-bounds (needed valid taps always covered)
  int hi_start = iclamp(2 * ho0 - 1, 0, INS - IWIN);
  int wi_start = iclamp(2 * wo0 - 1, 0, INS - IWIN);

  const float* ib = inter + (size_t)b * INS * INS * CH;

#if defined(__has_builtin) && __has_builtin(__builtin_amdgcn_tensor_load_to_lds)
  if (wave == 0) {
    // ---- Tensor DMA descriptor (ISA 8.3-8.6), 3D tile: c(32) x wi(36) x hi(36)
    unsigned long long gaddr =
        (unsigned long long)(uintptr_t)(ib + ((size_t)hi_start * INS + wi_start) * CH +
                                        cb * 32);
    unsigned int lds_off = (unsigned int)(uintptr_t)itile;  // low 32b = LDS byte offset

    u32x4 g0;
    g0.x = 1u;                               // count=1 (valid), user mode, no gather
    g0.y = lds_off;                          // lds_addr (bytes)
    g0.z = (unsigned int)gaddr;              // global_addr[31:0]
    g0.w = (unsigned int)((gaddr >> 32) & 0x1FFFFFFu) | (2u << 30);  // addr[56:32]|type=2

    i32x8 g1;
    g1.s0 = (int)((2u << 16) | (1u << 20) | (4u << 22) | (7u << 25));
    //            data_size=4B | pad_enable | pad_interval: 32 DW | pad_amount: 8 DW
    g1.s1 = (int)(32u << 16);         // tensor_dim0 = 32 (bits 79:48, low half)
    g1.s2 = (int)(36u << 16);         // tensor_dim0 hi=0 | tensor_dim1 = 36 (low half)
    g1.s3 = (int)(32u << 16);         // tensor_dim1 hi=0 | tile_dim0 = 32
    g1.s4 = (int)(36u | (36u << 16)); // tile_dim1 = 36 | tile_dim2 = 36
    g1.s5 = 64;                       // tensor_dim0_stride = 64 elements (channel stride)
    g1.s6 = (int)(16384u << 16);      // stride0 hi=0 | tensor_dim1_stride low16 = 16384
    g1.s7 = 0;                        // tensor_dim1_stride hi = 0

    i32x4 g2;
    g2.x = 36; g2.y = 1; g2.z = 0; g2.w = 0;  // tensor_dim2=36; dim3 unused (tile_dim3=0)
    i32x4 g3;
    g3.x = 0; g3.y = 0; g3.z = 0; g3.w = 0;

    i32x8 gx = {0, 0, 0, 0, 0, 0, 0, 0};  // extra arg of 6-arg builtin (unused)

    __builtin_amdgcn_tensor_load_to_lds(g0, g1, g2, g3, gx, 0);
    __builtin_amdgcn_s_wait_tensorcnt(0);
  }
#else
  // fallback: cooperative copy with identical LDS layout
  for (int e = threadIdx.x; e < IWIN * IWIN * 32; e += 256) {
    int c = e & 31;
    int l = e >> 5;
    int wi = l % IWIN, hi = l / IWIN;
    itile[l * ILINE + c] =
        ib[((size_t)(hi_start + hi) * INS + wi_start + wi) * CH + cb * 32 + c];
  }
#endif
  __syncthreads();

  // ---- stage 1: W-direction downsample (B from LDS itile) -> tbuf ----
  for (int task = wave; task < IWIN * 2; task += 8) {  // 72 tasks, 9/wave, uniform
    int hi_l = task >> 1;
    int c_off = (task & 1) * 16 + ln;
    int li = iclamp(2 * ho0 - 1 + hi_l - hi_start, 0, IWIN - 1);
    const float* irow = itile + li * (IWIN * ILINE);
    v8f acc = {};
#pragma unroll
    for (int kb = 0; kb < 9; ++kb) {
      int wib = 2 * wo0 - 1 + 4 * kb + 2 * grp;
      v2f a, bm;
      a.x = wdown(wo0 + ln, wib, OUTS, INS);
      a.y = wdown(wo0 + ln, wib + 1, OUTS, INS);
      int wl0 = iclamp(wib - wi_start, 0, IWIN - 1);
      int wl1 = iclamp(wib + 1 - wi_start, 0, IWIN - 1);
      bm.x = irow[wl0 * ILINE + c_off];
      bm.y = irow[wl1 * ILINE + c_off];
      acc = __builtin_amdgcn_wmma_f32_16x16x4_f32(false, a, false, bm, (short)0, acc,
                                                  false, false);
    }
    float* dst = &tbuf[hi_l * THSTR + c_off];
#pragma unroll
    for (int v = 0; v < 8; ++v) dst[(v + 8 * grp) * TWSTR] = acc[v];
  }

  __syncthreads();

  // ---- stage 2: H-direction downsample from tbuf, add bias, store ----
  float* ob = out + (size_t)b * OUTS * OUTS * CH;
  for (int task = wave; task < 32; task += 8) {  // 4/wave, uniform
    int wo_l = task >> 1;
    int c_off = (task & 1) * 16 + ln;
    v8f acc = {};
#pragma unroll
    for (int kb = 0; kb < 9; ++kb) {
      int hl = 4 * kb + 2 * grp;  // local hi row; max touched = 35 < IWIN
      int hg = 2 * ho0 - 1 + hl;
      v2f a, bm;
      a.x = wdown(ho0 + ln, hg, OUTS, INS);
      a.y = wdown(ho0 + ln, hg + 1, OUTS, INS);
      const float* src = &tbuf[hl * THSTR + wo_l * TWSTR + c_off];
      bm.x = src[0];
      bm.y = src[THSTR];
      acc = __builtin_amdgcn_wmma_f32_16x16x4_f32(false, a, false, bm, (short)0, acc,
                                                  false, false);
    }
    int c = cb * 32 + c_off;
    float bv = bias[c];
#pragma unroll
    for (int v = 0; v < 8; ++v) {
      int ho = ho0 + v + 8 * grp;
      ob[((size_t)ho * OUTS + wo0 + wo_l) * CH + c] = acc[v] + bv;
    }
  }
}

// ---------------------------------------------------------------------------
extern "C" void kernel_launch(void* const* d_in, const int* in_sizes, int n_in,
                              void* d_out, int out_size, void* d_ws, size_t ws_size,
                              hipStream_t stream) {
  const float* x = (const float*)d_in[0];     // (8,256,256,64) fp32
  const float* bias = (const float*)d_in[1];  // (64,) fp32
  float* out = (float*)d_out;                 // (8,128,128,64) fp32
  float* inter = (float*)d_ws;                // 8*256*256*64 fp32 = 128 MiB scratch

  (void)in_sizes; (void)n_in; (void)out_size; (void)ws_size;

  int k1_blocks = (BATCH * 64 * INS * CH) / 256;  // 32768
  k1_up_lrelu_down<<<k1_blocks, 256, 0, stream>>>(x, inter);

  // 8 batches * 8x8 tiles * 2 channel halves = 1024 blocks, 288 KB LDS each
  size_t lds_bytes = (size_t)LDS_FLOATS * sizeof(float);
  k2_down2_wmma<<<BATCH * 8 * 8 * 2, 256, lds_bytes, stream>>>(inter, bias, out);
}